// GATEncoder_54202487275959
// MI455X (gfx1250) — compile-verified
//
#include <hip/hip_runtime.h>
#include <hip/hip_bf16.h>

typedef __attribute__((ext_vector_type(2))) float v2f;
typedef __attribute__((ext_vector_type(8))) float v8f;

#define NEG_SLOPE 0.2f

// ---------- helpers: order-preserving float<->uint encoding for atomicMax ----------
__device__ __forceinline__ unsigned encf(float x) {
  unsigned u = __float_as_uint(x);
  return (u & 0x80000000u) ? ~u : (u | 0x80000000u);
}
__device__ __forceinline__ float decf(unsigned u) {
  return (u & 0x80000000u) ? __uint_as_float(u ^ 0x80000000u) : __uint_as_float(~u);
}

// ---------- tiny utility kernels ----------
__global__ __launch_bounds__(256) void k_fill_f32(float* p, float v, int n) {
  int i = blockIdx.x * 256 + threadIdx.x;
  if (i < n) p[i] = v;
}
__global__ __launch_bounds__(256) void k_fill_u32(unsigned* p, unsigned v, int n) {
  int i = blockIdx.x * 256 + threadIdx.x;
  if (i < n) p[i] = v;
}
__global__ __launch_bounds__(256) void k_bias(float* acc, const float* __restrict__ b, int C, int total) {
  int i = blockIdx.x * 256 + threadIdx.x;
  if (i < total) acc[i] = b[i % C];
}
__global__ __launch_bounds__(256) void k_relu(float* p, int n) {
  int i = blockIdx.x * 256 + threadIdx.x;
  if (i < n) { float v = p[i]; p[i] = v > 0.f ? v : 0.f; }
}

// ---------- GEMM 1: X[N,128] @ W1[128,32] -> H[N,32], fp32 WMMA ----------
// 8 waves/block, 16 rows/wave -> 128 rows/block. W1 cached in LDS (16 KB).
__global__ __launch_bounds__(256) void k_gemm1(const float* __restrict__ X,
                                               const float* __restrict__ W,
                                               float* __restrict__ H, int N) {
  __shared__ float w[128 * 32];
  for (int i = threadIdx.x; i < 128 * 32; i += 256) w[i] = W[i];
  __syncthreads();

  const int wave = threadIdx.x >> 5;
  const int lane = threadIdx.x & 31;
  const int m = lane & 15;         // col-within-tile / row-of-A
  const int half = lane >> 4;      // selects K pair
  const int rowBase = blockIdx.x * 128 + wave * 16;

  int rowA = rowBase + m;          // clamp loads; stores are guarded
  if (rowA > N - 1) rowA = N - 1;
  const float* xr = X + (size_t)rowA * 128;

  v8f acc0 = {};
  v8f acc1 = {};
#pragma unroll 4
  for (int k0 = 0; k0 < 128; k0 += 4) {
    const int k = k0 + 2 * half;
    v2f a;  a[0] = xr[k];                 a[1] = xr[k + 1];
    v2f b0; b0[0] = w[k * 32 + m];        b0[1] = w[(k + 1) * 32 + m];
    v2f b1; b1[0] = w[k * 32 + 16 + m];   b1[1] = w[(k + 1) * 32 + 16 + m];
    acc0 = __builtin_amdgcn_wmma_f32_16x16x4_f32(false, a, false, b0, (short)0, acc0, false, false);
    acc1 = __builtin_amdgcn_wmma_f32_16x16x4_f32(false, a, false, b1, (short)0, acc1, false, false);
  }
#pragma unroll
  for (int r = 0; r < 8; ++r) {
    int row = rowBase + r + 8 * half;   // D layout: lanes 0-15 -> M=r, lanes 16-31 -> M=r+8
    if (row < N) {
      H[(size_t)row * 32 + m]      = acc0[r];
      H[(size_t)row * 32 + 16 + m] = acc1[r];
    }
  }
}

// ---------- GEMM 2 (fused): H1[N,32] @ Wmu[32,16] and @ Wls[32,16] ----------
__global__ __launch_bounds__(256) void k_gemm2(const float* __restrict__ H1,
                                               const float* __restrict__ Wmu,
                                               const float* __restrict__ Wls,
                                               float* __restrict__ Hmu,
                                               float* __restrict__ Hls, int N) {
  __shared__ float wm[32 * 16];
  __shared__ float wl[32 * 16];
  for (int i = threadIdx.x; i < 512; i += 256) { wm[i] = Wmu[i]; wl[i] = Wls[i]; }
  __syncthreads();

  const int wave = threadIdx.x >> 5;
  const int lane = threadIdx.x & 31;
  const int m = lane & 15;
  const int half = lane >> 4;
  const int rowBase = blockIdx.x * 128 + wave * 16;

  int rowA = rowBase + m;
  if (rowA > N - 1) rowA = N - 1;
  const float* hr = H1 + (size_t)rowA * 32;

  v8f am = {};
  v8f al = {};
#pragma unroll
  for (int k0 = 0; k0 < 32; k0 += 4) {
    const int k = k0 + 2 * half;
    v2f a;  a[0] = hr[k];             a[1] = hr[k + 1];
    v2f bm; bm[0] = wm[k * 16 + m];   bm[1] = wm[(k + 1) * 16 + m];
    v2f bl; bl[0] = wl[k * 16 + m];   bl[1] = wl[(k + 1) * 16 + m];
    am = __builtin_amdgcn_wmma_f32_16x16x4_f32(false, a, false, bm, (short)0, am, false, false);
    al = __builtin_amdgcn_wmma_f32_16x16x4_f32(false, a, false, bl, (short)0, al, false, false);
  }
#pragma unroll
  for (int r = 0; r < 8; ++r) {
    int row = rowBase + r + 8 * half;
    if (row < N) {
      Hmu[(size_t)row * 16 + m] = am[r];
      Hls[(size_t)row * 16 + m] = al[r];
    }
  }
}

// ---------- per-row attention logits: s=h.a_src, d=h.a_dst ----------
__global__ __launch_bounds__(256) void k_alpha(const float* __restrict__ H, int C,
                                               const float* __restrict__ a_src,
                                               const float* __restrict__ a_dst,
                                               float* __restrict__ out_s,
                                               float* __restrict__ out_d, int N) {
  int i = blockIdx.x * 256 + threadIdx.x;
  if (i >= N) return;
  const float* h = H + (size_t)i * C;
  float s = 0.f, d = 0.f;
  for (int c = 0; c < C; ++c) { float v = h[c]; s += v * a_src[c]; d += v * a_dst[c]; }
  out_s[i] = s;
  out_d[i] = d;
}

// ---------- edge index decode (concat of edge_index and self-loops) ----------
__device__ __forceinline__ void edge_sd(const long long* __restrict__ ei, int Eraw,
                                        int e, int& s, int& d) {
  if (e < Eraw) { s = (int)ei[e]; d = (int)ei[(size_t)Eraw + e]; }
  else          { s = d = e - Eraw; }
}

// ---------- edge pass 1: e = leaky_relu(as[src]+ad[dst]); segment max ----------
__global__ __launch_bounds__(256) void k_edge_max1(const long long* __restrict__ ei, int Eraw, int Etot,
                                                   const float* __restrict__ as_, const float* __restrict__ ad_,
                                                   float* __restrict__ e_out, unsigned* __restrict__ maxEnc) {
  int e = blockIdx.x * 256 + threadIdx.x;
  if (e >= Etot) return;
  int s, d; edge_sd(ei, Eraw, e, s, d);
  float v = as_[s] + ad_[d];
  v = v > 0.f ? v : NEG_SLOPE * v;
  e_out[e] = v;
  atomicMax(&maxEnc[d], encf(v));
}

__global__ __launch_bounds__(256) void k_edge_max2(const long long* __restrict__ ei, int Eraw, int Etot,
                                                   const float* __restrict__ as0, const float* __restrict__ ad0,
                                                   const float* __restrict__ as1, const float* __restrict__ ad1,
                                                   float* __restrict__ e0, float* __restrict__ e1,
                                                   unsigned* __restrict__ mx0, unsigned* __restrict__ mx1) {
  int e = blockIdx.x * 256 + threadIdx.x;
  if (e >= Etot) return;
  int s, d; edge_sd(ei, Eraw, e, s, d);
  float v0 = as0[s] + ad0[d]; v0 = v0 > 0.f ? v0 : NEG_SLOPE * v0;
  float v1 = as1[s] + ad1[d]; v1 = v1 > 0.f ? v1 : NEG_SLOPE * v1;
  e0[e] = v0; e1[e] = v1;
  atomicMax(&mx0[d], encf(v0));
  atomicMax(&mx1[d], encf(v1));
}

// ---------- edge pass 2: ex = exp(e - max[dst]); segment sum ----------
__global__ __launch_bounds__(256) void k_edge_sum1(const long long* __restrict__ ei, int Eraw, int Etot,
                                                   float* __restrict__ e_io,
                                                   const unsigned* __restrict__ maxEnc,
                                                   float* __restrict__ sum) {
  int e = blockIdx.x * 256 + threadIdx.x;
  if (e >= Etot) return;
  int d = (e < Eraw) ? (int)ei[(size_t)Eraw + e] : (e - Eraw);
  float ex = expf(e_io[e] - decf(maxEnc[d]));
  e_io[e] = ex;
  atomicAdd(&sum[d], ex);
}

__global__ __launch_bounds__(256) void k_edge_sum2(const long long* __restrict__ ei, int Eraw, int Etot,
                                                   float* __restrict__ e0, float* __restrict__ e1,
                                                   const unsigned* __restrict__ mx0, const unsigned* __restrict__ mx1,
                                                   float* __restrict__ sm0, float* __restrict__ sm1) {
  int e = blockIdx.x * 256 + threadIdx.x;
  if (e >= Etot) return;
  int d = (e < Eraw) ? (int)ei[(size_t)Eraw + e] : (e - Eraw);
  float x0 = expf(e0[e] - decf(mx0[d]));
  float x1 = expf(e1[e] - decf(mx1[d]));
  e0[e] = x0; e1[e] = x1;
  atomicAdd(&sm0[d], x0);
  atomicAdd(&sm1[d], x1);
}

// ---------- edge pass 3: acc[dst] += h[src] * alpha ----------
__global__ __launch_bounds__(256) void k_edge_scatter1(const long long* __restrict__ ei, int Eraw, int Etot,
                                                       const float* __restrict__ ex, const float* __restrict__ sum,
                                                       const float* __restrict__ H, float* __restrict__ acc, int C) {
  int e = blockIdx.x * 256 + threadIdx.x;
  if (e >= Etot) return;
  int s, d; edge_sd(ei, Eraw, e, s, d);
  float alpha = ex[e] / sum[d];
  const float* hs = H + (size_t)s * C;
  float* ao = acc + (size_t)d * C;
  for (int c = 0; c < C; c += 4) {
    float4 v = *(const float4*)(hs + c);
    atomicAdd(ao + c + 0, v.x * alpha);
    atomicAdd(ao + c + 1, v.y * alpha);
    atomicAdd(ao + c + 2, v.z * alpha);
    atomicAdd(ao + c + 3, v.w * alpha);
  }
}

__global__ __launch_bounds__(256) void k_edge_scatter2(const long long* __restrict__ ei, int Eraw, int Etot,
                                                       const float* __restrict__ ex0, const float* __restrict__ sm0,
                                                       const float* __restrict__ ex1, const float* __restrict__ sm1,
                                                       const float* __restrict__ H0, const float* __restrict__ H1,
                                                       float* __restrict__ a0, float* __restrict__ a1) {
  int e = blockIdx.x * 256 + threadIdx.x;
  if (e >= Etot) return;
  int s, d; edge_sd(ei, Eraw, e, s, d);
  float al0 = ex0[e] / sm0[d];
  float al1 = ex1[e] / sm1[d];
  const float* h0 = H0 + (size_t)s * 16;
  const float* h1 = H1 + (size_t)s * 16;
  float* o0 = a0 + (size_t)d * 16;
  float* o1 = a1 + (size_t)d * 16;
#pragma unroll
  for (int c = 0; c < 16; c += 4) {
    float4 v0 = *(const float4*)(h0 + c);
    float4 v1 = *(const float4*)(h1 + c);
    atomicAdd(o0 + c + 0, v0.x * al0); atomicAdd(o0 + c + 1, v0.y * al0);
    atomicAdd(o0 + c + 2, v0.z * al0); atomicAdd(o0 + c + 3, v0.w * al0);
    atomicAdd(o1 + c + 0, v1.x * al1); atomicAdd(o1 + c + 1, v1.y * al1);
    atomicAdd(o1 + c + 2, v1.z * al1); atomicAdd(o1 + c + 3, v1.w * al1);
  }
}

// ---------------------------------------------------------------------------
extern "C" void kernel_launch(void* const* d_in, const int* in_sizes, int n_in,
                              void* d_out, int out_size, void* d_ws, size_t ws_size,
                              hipStream_t stream) {
  const float*     x    = (const float*)d_in[0];
  const long long* ei   = (const long long*)d_in[1];   // int64 edge_index [2, Eraw]
  const float* W1   = (const float*)d_in[2];
  const float* a1s  = (const float*)d_in[3];
  const float* a1d  = (const float*)d_in[4];
  const float* b1   = (const float*)d_in[5];
  const float* Wmu  = (const float*)d_in[6];
  const float* amus = (const float*)d_in[7];
  const float* amud = (const float*)d_in[8];
  const float* bmu  = (const float*)d_in[9];
  const float* Wls  = (const float*)d_in[10];
  const float* alss = (const float*)d_in[11];
  const float* alsd = (const float*)d_in[12];
  const float* bls  = (const float*)d_in[13];

  const int N    = in_sizes[0] / 128;
  const int Eraw = in_sizes[1] / 2;
  const int Etot = Eraw + N;

  // ---- workspace layout (floats) ----
  float* ws = (float*)d_ws;
  float* hbuf = ws;                    // N*32 : h1 pre-act; later h_mu (N*16) | h_ls (N*16)
  float* acc1 = hbuf + (size_t)N * 32; // N*32 : layer-1 accumulator -> h1 (post relu)
  float* eA   = acc1 + (size_t)N * 32; // Etot : e/ex buffer (layer1, then mu)
  float* eB   = eA + (size_t)Etot;     // Etot : e/ex buffer (ls)
  float* scal = eB + (size_t)Etot;     // 12 * N scalars
  float*    as1  = scal + 0 * (size_t)N;
  float*    ad1  = scal + 1 * (size_t)N;
  float*    asmu = scal + 2 * (size_t)N;
  float*    admu = scal + 3 * (size_t)N;
  float*    asls = scal + 4 * (size_t)N;
  float*    adls = scal + 5 * (size_t)N;
  float*    sum1 = scal + 6 * (size_t)N;
  float*    summ = scal + 7 * (size_t)N;
  float*    suml = scal + 8 * (size_t)N;
  unsigned* max1 = (unsigned*)(scal + 9 * (size_t)N);
  unsigned* maxm = (unsigned*)(scal + 10 * (size_t)N);
  unsigned* maxl = (unsigned*)(scal + 11 * (size_t)N);

  float* acc_mu = (float*)d_out;              // [N,16] mu accumulator (init = bias)
  float* acc_ls = acc_mu + (size_t)N * 16;    // [N,16] logstd accumulator

  const int gN    = (N + 255) / 256;
  const int gN32  = (N * 32 + 255) / 256;
  const int gN16  = (N * 16 + 255) / 256;
  const int gE    = (Etot + 255) / 256;
  const int gRows = (N + 127) / 128;

  // ================= Layer 1: 128 -> 32, relu =================
  k_fill_f32<<<gN, 256, 0, stream>>>(sum1, 0.f, N);
  k_fill_u32<<<gN, 256, 0, stream>>>(max1, 0u, N);
  k_bias<<<gN32, 256, 0, stream>>>(acc1, b1, 32, N * 32);

  k_gemm1<<<gRows, 256, 0, stream>>>(x, W1, hbuf, N);
  k_alpha<<<gN, 256, 0, stream>>>(hbuf, 32, a1s, a1d, as1, ad1, N);

  k_edge_max1<<<gE, 256, 0, stream>>>(ei, Eraw, Etot, as1, ad1, eA, max1);
  k_edge_sum1<<<gE, 256, 0, stream>>>(ei, Eraw, Etot, eA, max1, sum1);
  k_edge_scatter1<<<gE, 256, 0, stream>>>(ei, Eraw, Etot, eA, sum1, hbuf, acc1, 32);
  k_relu<<<gN32, 256, 0, stream>>>(acc1, N * 32);   // acc1 is now h1

  // ================= Layer 2: 32 -> 16 (mu & logstd fused) =================
  k_fill_f32<<<gN, 256, 0, stream>>>(summ, 0.f, N);
  k_fill_f32<<<gN, 256, 0, stream>>>(suml, 0.f, N);
  k_fill_u32<<<gN, 256, 0, stream>>>(maxm, 0u, N);
  k_fill_u32<<<gN, 256, 0, stream>>>(maxl, 0u, N);
  k_bias<<<gN16, 256, 0, stream>>>(acc_mu, bmu, 16, N * 16);
  k_bias<<<gN16, 256, 0, stream>>>(acc_ls, bls, 16, N * 16);

  float* hmu = hbuf;                  // reuse
  float* hls = hbuf + (size_t)N * 16;
  k_gemm2<<<gRows, 256, 0, stream>>>(acc1, Wmu, Wls, hmu, hls, N);
  k_alpha<<<gN, 256, 0, stream>>>(hmu, 16, amus, amud, asmu, admu, N);
  k_alpha<<<gN, 256, 0, stream>>>(hls, 16, alss, alsd, asls, adls, N);

  k_edge_max2<<<gE, 256, 0, stream>>>(ei, Eraw, Etot, asmu, admu, asls, adls, eA, eB, maxm, maxl);
  k_edge_sum2<<<gE, 256, 0, stream>>>(ei, Eraw, Etot, eA, eB, maxm, maxl, summ, suml);
  k_edge_scatter2<<<gE, 256, 0, stream>>>(ei, Eraw, Etot, eA, summ, eB, suml, hmu, hls, acc_mu, acc_ls);
}